// CTRGC_16209206575445
// MI455X (gfx1250) — compile-verified
//
#include <hip/hip_runtime.h>
#include <hip/hip_bf16.h>

typedef float v2f __attribute__((ext_vector_type(2)));
typedef float v8f __attribute__((ext_vector_type(8)));

// dims: N=64, C=128, T=64, V=25, O=128, R=16, CH=64, TV=1600

// ---------------- K1: fused reductions over x ----------------
// xm[n,c,t] = mean_v x ; xt[n,c,v] = mean_t x ; g0[n,c] = mean_{t,v} x
__global__ __launch_bounds__(128) void k1_reduce(const float* __restrict__ x,
        float* __restrict__ xm, float* __restrict__ xt, float* __restrict__ g0) {
    __shared__ float s[1600];
    __shared__ float rowsum[64];
    int nc = blockIdx.x;                       // 0..8191  (n*128 + c)
    const float* xp = x + (size_t)nc * 1600;
    for (int i = threadIdx.x; i < 1600; i += 128) s[i] = xp[i];
    __syncthreads();
    int tid = threadIdx.x;
    if (tid < 64) {
        float acc = 0.f;
        #pragma unroll
        for (int v = 0; v < 25; ++v) acc += s[tid * 25 + v];
        rowsum[tid] = acc;
        xm[nc * 64 + tid] = acc * (1.0f / 25.0f);
    } else if (tid < 89) {
        int v = tid - 64;
        float acc = 0.f;
        for (int t = 0; t < 64; ++t) acc += s[t * 25 + v];
        xt[nc * 25 + v] = acc * (1.0f / 64.0f);
    }
    __syncthreads();
    if (tid == 0) {
        float tot = 0.f;
        for (int t = 0; t < 64; ++t) tot += rowsum[t];
        g0[nc] = tot * (1.0f / 1600.0f);
    }
}

// ---------------- K0: transpose tada_w -> wt[c][u] ----------------
__global__ __launch_bounds__(256) void k0_wt(const float* __restrict__ w,
                                             float* __restrict__ wt) {
    int i = blockIdx.x * 256 + threadIdx.x;    // 16384 total
    int u = i >> 7, c = i & 127;
    wt[c * 128 + u] = w[i];
}

// ---------------- K2a: g[n,o] = g0[n,:] @ rf_g_w[o,:] + b ----------------
__global__ __launch_bounds__(128) void k2_g(const float* __restrict__ g0,
        const float* __restrict__ w, const float* __restrict__ b,
        float* __restrict__ g) {
    __shared__ float s[128];
    int n = blockIdx.x;
    s[threadIdx.x] = g0[n * 128 + threadIdx.x];
    __syncthreads();
    int o = threadIdx.x;
    float acc = b[o];
    const float* wp = w + o * 128;
    for (int c = 0; c < 128; ++c) acc += s[c] * wp[c];
    g[n * 128 + o] = acc;
}

// ---------------- K2b: a = relu(BN(conv1d(xm + g, rf_a_w) + bias)) ----------------
__global__ __launch_bounds__(256) void k2_conva(const float* __restrict__ xm,
        const float* __restrict__ g, const float* __restrict__ w,
        const float* __restrict__ bias, const float* __restrict__ gamma,
        const float* __restrict__ beta, float* __restrict__ aout) {
    __shared__ float xa[128 * 64];
    int n = blockIdx.x;
    for (int i = threadIdx.x; i < 128 * 64; i += 256) {
        int c = i >> 6;
        xa[i] = xm[(size_t)n * 8192 + i] + g[n * 128 + c];
    }
    __syncthreads();
    const float inv = rsqrtf(1.0f + 1e-5f);
    for (int idx = threadIdx.x; idx < 64 * 64; idx += 256) {
        int ch = idx >> 6, t = idx & 63;
        float acc = bias[ch];
        const float* wp = w + ch * 128 * 3;        // (CH, C, 3)
        for (int c = 0; c < 128; ++c) {
            const float* xr = xa + c * 64;
            float xl = (t > 0) ? xr[t - 1] : 0.f;
            float xh = (t < 63) ? xr[t + 1] : 0.f;
            acc += wp[c * 3 + 0] * xl + wp[c * 3 + 1] * xr[t] + wp[c * 3 + 2] * xh;
        }
        float val = acc * (gamma[ch] * inv) + beta[ch];
        aout[(size_t)n * 4096 + idx] = fmaxf(val, 0.f);
    }
}

// ---------------- K2c: alpha_rf = conv1d(a, rf_b_w) + 1 ----------------
__global__ __launch_bounds__(256) void k2_convb(const float* __restrict__ a,
        const float* __restrict__ w, float* __restrict__ arf) {
    __shared__ float ab[64 * 64];
    int n = blockIdx.x;
    for (int i = threadIdx.x; i < 4096; i += 256) ab[i] = a[(size_t)n * 4096 + i];
    __syncthreads();
    for (int idx = threadIdx.x; idx < 128 * 64; idx += 256) {
        int o = idx >> 6, t = idx & 63;
        float acc = 1.0f;
        const float* wp = w + o * 64 * 3;          // (O, CH, 3)
        for (int c = 0; c < 64; ++c) {
            const float* xr = ab + c * 64;
            float xl = (t > 0) ? xr[t - 1] : 0.f;
            float xh = (t < 63) ? xr[t + 1] : 0.f;
            acc += wp[c * 3 + 0] * xl + wp[c * 3 + 1] * xr[t] + wp[c * 3 + 2] * xh;
        }
        arf[(size_t)n * 8192 + idx] = acc;
    }
}

// ---------------- K2d: x1/x2 = conv1/conv2 applied to xt ----------------
__global__ __launch_bounds__(128) void k2_x12(const float* __restrict__ xt,
        const float* __restrict__ w1, const float* __restrict__ b1,
        const float* __restrict__ w2, const float* __restrict__ b2,
        float* __restrict__ x1, float* __restrict__ x2) {
    __shared__ float s[128 * 25];
    int n = blockIdx.x;
    for (int i = threadIdx.x; i < 3200; i += 128) s[i] = xt[(size_t)n * 3200 + i];
    __syncthreads();
    for (int idx = threadIdx.x; idx < 800; idx += 128) {
        int which = idx / 400;
        int rv = idx % 400;
        int r = rv / 25, v = rv % 25;
        const float* w = which ? w2 : w1;
        float acc = which ? b2[r] : b1[r];
        const float* wp = w + r * 128;
        for (int c = 0; c < 128; ++c) acc += wp[c] * s[c * 25 + v];
        (which ? x2 : x1)[n * 400 + rv] = acc;
    }
}

// ---------------- K3: m = alpha*(conv4_w @ tanh(x1-x2) + b4) + A ----------------
__global__ __launch_bounds__(256) void k3_m(const float* __restrict__ x1,
        const float* __restrict__ x2, const float* __restrict__ w4,
        const float* __restrict__ b4, const float* __restrict__ A,
        const int* __restrict__ alpha, float* __restrict__ m) {
    __shared__ float diff[16 * 625];
    int n = blockIdx.x >> 3;
    int oc = blockIdx.x & 7;                   // 16 output channels per block
    float al = (float)alpha[0];
    for (int idx = threadIdx.x; idx < 16 * 625; idx += 256) {
        int r = idx / 625, uv = idx % 625;
        int u = uv / 25, v = uv % 25;
        diff[idx] = tanhf(x1[n * 400 + r * 25 + u] - x2[n * 400 + r * 25 + v]);
    }
    __syncthreads();
    for (int idx = threadIdx.x; idx < 16 * 625; idx += 256) {
        int ol = idx / 625, uv = idx % 625;
        int o = oc * 16 + ol;
        float acc = 0.f;
        const float* wp = w4 + o * 16;
        #pragma unroll
        for (int r = 0; r < 16; ++r) acc += wp[r] * diff[r * 625 + uv];
        m[((size_t)n * 128 + o) * 625 + uv] = (acc + b4[o]) * al + A[uv];
    }
}

// ---------------- K4: y[n,u,col] = arf[n,u,t] * (W @ x[n])  via WMMA f32 ----------------
// block = (n, 64-column group). 8 waves = 8 u-tiles; each wave does 4 col-tiles
// sharing one A fragment per K-step (4 independent accumulator chains).
__global__ __launch_bounds__(256) void k4_gemm(const float* __restrict__ x,
        const float* __restrict__ wt, const float* __restrict__ arf,
        float* __restrict__ y) {
    __shared__ float xb[128 * 64];             // [k(c)][col]
    int n = blockIdx.x / 25;
    int cg = blockIdx.x % 25;
    int colbase = cg * 64;
    const float* xn = x + (size_t)n * 128 * 1600;
    for (int i = threadIdx.x; i < 128 * 64; i += 256) {
        int c = i >> 6, col = i & 63;
        xb[i] = xn[(size_t)c * 1600 + colbase + col];
    }
    __syncthreads();
    int wid = threadIdx.x >> 5;                // wave 0..7 -> u tile
    int lane = threadIdx.x & 31;
    int ml = lane & 15, h = lane >> 4;         // A row / B col ; lane half
    int u0 = wid * 16;
    v8f acc0 = {}, acc1 = {}, acc2 = {}, acc3 = {};
    const float* wtp = wt + u0 + ml;           // W^T[c][u]: lane-contiguous
    #pragma unroll 4
    for (int k = 0; k < 128; k += 4) {
        int kk = k + 2 * h;                    // frag element j holds K = kk + j
        v2f a;
        a.x = wtp[(size_t)kk * 128];
        a.y = wtp[(size_t)(kk + 1) * 128];
        const float* xr = xb + kk * 64 + ml;
        v2f b0, b1, b2, b3;
        b0.x = xr[0];      b0.y = xr[64];
        b1.x = xr[16];     b1.y = xr[80];
        b2.x = xr[32];     b2.y = xr[96];
        b3.x = xr[48];     b3.y = xr[112];
        acc0 = __builtin_amdgcn_wmma_f32_16x16x4_f32(false, a, false, b0, (short)0, acc0, false, false);
        acc1 = __builtin_amdgcn_wmma_f32_16x16x4_f32(false, a, false, b1, (short)0, acc1, false, false);
        acc2 = __builtin_amdgcn_wmma_f32_16x16x4_f32(false, a, false, b2, (short)0, acc2, false, false);
        acc3 = __builtin_amdgcn_wmma_f32_16x16x4_f32(false, a, false, b3, (short)0, acc3, false, false);
    }
    float* yn = y + (size_t)n * 128 * 1600;
    #pragma unroll
    for (int ct = 0; ct < 4; ++ct) {
        v8f acc = (ct == 0) ? acc0 : (ct == 1) ? acc1 : (ct == 2) ? acc2 : acc3;
        int col = colbase + ct * 16 + ml;
        int t = col / 25;
        #pragma unroll
        for (int j = 0; j < 8; ++j) {
            int u = u0 + j + 8 * h;            // C/D layout: row = j + 8*laneHalf
            yn[(size_t)u * 1600 + col] = acc[j] * arf[((size_t)n * 128 + u) * 64 + t];
        }
    }
}

// ---------------- K5: out[n,c,t,u] = sum_v m[n,c,u,v] * y[n,c,t,v]  via WMMA f32 ----------------
// block = one (n,c); Y and m^T staged in LDS with zero padding V:25->28.
__global__ __launch_bounds__(256) void k5_out(const float* __restrict__ y,
        const float* __restrict__ m, float* __restrict__ out) {
    __shared__ float yb[64 * 28];              // [t][v_pad]
    __shared__ float mt[28 * 32];              // [v_pad][u_pad]
    int nc = blockIdx.x;                       // 0..8191
    for (int i = threadIdx.x; i < 64 * 28; i += 256) yb[i] = 0.f;
    for (int i = threadIdx.x; i < 28 * 32; i += 256) mt[i] = 0.f;
    __syncthreads();
    const float* yp = y + (size_t)nc * 1600;
    const float* mp = m + (size_t)nc * 625;
    for (int i = threadIdx.x; i < 1600; i += 256) {
        int t = i / 25, v = i % 25;
        yb[t * 28 + v] = yp[i];
    }
    for (int i = threadIdx.x; i < 625; i += 256) {
        int u = i / 25, v = i % 25;
        mt[v * 32 + u] = mp[i];                // transpose: B[v][u] = m[u][v]
    }
    __syncthreads();
    int wid = threadIdx.x >> 5, lane = threadIdx.x & 31;
    int ml = lane & 15, h = lane >> 4;
    int t0 = (wid & 3) * 16, u0 = (wid >> 2) * 16;
    v8f acc = {};
    #pragma unroll
    for (int k = 0; k < 28; k += 4) {
        int kk = k + 2 * h;
        v2f a, b;
        a.x = yb[(t0 + ml) * 28 + kk];
        a.y = yb[(t0 + ml) * 28 + kk + 1];
        b.x = mt[kk * 32 + u0 + ml];
        b.y = mt[(kk + 1) * 32 + u0 + ml];
        acc = __builtin_amdgcn_wmma_f32_16x16x4_f32(false, a, false, b, (short)0, acc, false, false);
    }
    int u = u0 + ml;
    if (u < 25) {
        float* op = out + (size_t)nc * 1600;
        #pragma unroll
        for (int j = 0; j < 8; ++j) {
            int t = t0 + j + 8 * h;
            op[t * 25 + u] = acc[j];
        }
    }
}

extern "C" void kernel_launch(void* const* d_in, const int* in_sizes, int n_in,
                              void* d_out, int out_size, void* d_ws, size_t ws_size,
                              hipStream_t stream) {
    const float* x        = (const float*)d_in[0];
    const float* A        = (const float*)d_in[1];
    const float* conv1_w  = (const float*)d_in[2];
    const float* conv1_b  = (const float*)d_in[3];
    const float* conv2_w  = (const float*)d_in[4];
    const float* conv2_b  = (const float*)d_in[5];
    const float* conv4_w  = (const float*)d_in[6];
    const float* conv4_b  = (const float*)d_in[7];
    const float* rf_g_w   = (const float*)d_in[8];
    const float* rf_g_b   = (const float*)d_in[9];
    const float* rf_a_w   = (const float*)d_in[10];
    const float* rf_a_b   = (const float*)d_in[11];
    const float* bn_gamma = (const float*)d_in[12];
    const float* bn_beta  = (const float*)d_in[13];
    const float* rf_b_w   = (const float*)d_in[14];
    const float* tada_w   = (const float*)d_in[15];
    const int*   alpha    = (const int*)d_in[16];
    float* out = (float*)d_out;
    float* ws  = (float*)d_ws;

    float* yws   = ws;                   // 13,107,200 floats (N*O*T*V)
    float* mws   = yws   + 13107200;     //  5,120,000 (N*O*V*V)
    float* xmws  = mws   + 5120000;      //    524,288 (N*C*T)
    float* xtws  = xmws  + 524288;       //    204,800 (N*C*V)
    float* g0ws  = xtws  + 204800;       //      8,192
    float* gws   = g0ws  + 8192;         //      8,192
    float* x1ws  = gws   + 8192;         //     25,600
    float* x2ws  = x1ws  + 25600;        //     25,600
    float* aws   = x2ws  + 25600;        //    262,144 (N*CH*T)
    float* arfws = aws   + 262144;       //    524,288 (N*O*T)
    float* wtws  = arfws + 524288;       //     16,384 (C*O transposed W)

    k1_reduce<<<8192, 128, 0, stream>>>(x, xmws, xtws, g0ws);
    k0_wt<<<64, 256, 0, stream>>>(tada_w, wtws);
    k2_g<<<64, 128, 0, stream>>>(g0ws, rf_g_w, rf_g_b, gws);
    k2_conva<<<64, 256, 0, stream>>>(xmws, gws, rf_a_w, rf_a_b, bn_gamma, bn_beta, aws);
    k2_convb<<<64, 256, 0, stream>>>(aws, rf_b_w, arfws);
    k2_x12<<<64, 128, 0, stream>>>(xtws, conv1_w, conv1_b, conv2_w, conv2_b, x1ws, x2ws);
    k3_m<<<512, 256, 0, stream>>>(x1ws, x2ws, conv4_w, conv4_b, A, alpha, mws);
    k4_gemm<<<1600, 256, 0, stream>>>(x, wtws, arfws, yws);
    k5_out<<<8192, 256, 0, stream>>>(yws, mws, out);
}